// SAModule_34849364640189
// MI455X (gfx1250) — compile-verified
//
#include <hip/hip_runtime.h>
#include <hip/hip_bf16.h>
#include <cstdint>

// ---------------- problem constants (match reference) ----------------
#define N_PTS   1048576
#define N_FEAT  64
#define GRIDSZ  0.05f

// voxel hash space capacity: actual dims ~ 21*21*21*16 = 148176 << 2^19
#define VOXCAP  (1 << 19)

// scan configuration: VOXCAP = SCAN_NBLK * SCAN_TILE
#define SCAN_BLOCK 256
#define SCAN_EPT   8
#define SCAN_TILE  (SCAN_BLOCK * SCAN_EPT)   // 2048
#define SCAN_NBLK  (VOXCAP / SCAN_TILE)      // 256

// header layout (ints) in workspace
#define HDR_MINKEY 0   // 4 sortable-uint min keys (x,y,z,batch)
#define HDR_MAXKEY 4   // 4 sortable-uint max keys
#define HDR_START  8   // 4 floats (bit-cast) start coords
#define HDR_STRIDE 12  // 4 int strides (s0=1)
#define HDR_T      16  // total voxels (capped)
#define HDR_K      17  // number of occupied voxels (clusters)
#define HDR_NTOT   18  // total points (scan check)

typedef unsigned v4u __attribute__((ext_vector_type(4)));
typedef unsigned v8u __attribute__((ext_vector_type(8)));

// ---------------- helpers ----------------
__device__ __forceinline__ unsigned fkey(float f) {
  unsigned u = __float_as_uint(f);
  return (u & 0x80000000u) ? ~u : (u | 0x80000000u);   // order-preserving uint key
}
__device__ __forceinline__ float fdec(unsigned u) {
  return (u & 0x80000000u) ? __uint_as_float(u ^ 0x80000000u)
                           : __uint_as_float(~u);
}

// ---------------- kernels ----------------
__global__ void k_init(float* __restrict__ out, int* __restrict__ hdr,
                       int* __restrict__ occ, int* __restrict__ pref,
                       int* __restrict__ counts, int* __restrict__ cstart,
                       int* __restrict__ cursor) {
  const float4 z4 = make_float4(0.f, 0.f, 0.f, 0.f);
  const int stride = gridDim.x * blockDim.x;
  for (int i = blockIdx.x * blockDim.x + threadIdx.x;
       i < N_PTS * (N_FEAT / 4); i += stride) {
    reinterpret_cast<float4*>(out)[i] = z4;                   // x region <- 0
    if (i < N_PTS * 3) out[(size_t)N_PTS * N_FEAT + i] = 0.f; // pos
    if (i < N_PTS)     out[(size_t)N_PTS * 67 + i] = 0.f;     // batch
    if (i < VOXCAP) {
      occ[i] = 0; pref[i] = 0; counts[i] = 0; cstart[i] = 0; cursor[i] = 0;
    }
    if (i < 4)           hdr[HDR_MINKEY + i] = (int)0xFFFFFFFFu;
    if (i >= 4 && i < 8) hdr[i] = 0;                          // max keys
  }
}

__global__ void k_minmax(const float* __restrict__ pos, const int* __restrict__ batch,
                         unsigned* __restrict__ hdr) {
  unsigned mn[4] = {0xFFFFFFFFu, 0xFFFFFFFFu, 0xFFFFFFFFu, 0xFFFFFFFFu};
  unsigned mx[4] = {0u, 0u, 0u, 0u};
  const int stride = gridDim.x * blockDim.x;
  for (int p = blockIdx.x * blockDim.x + threadIdx.x; p < N_PTS; p += stride) {
    const float c[4] = { pos[(size_t)p * 3 + 0], pos[(size_t)p * 3 + 1],
                         pos[(size_t)p * 3 + 2], (float)batch[p] };
#pragma unroll
    for (int d = 0; d < 4; ++d) {
      const unsigned k = fkey(c[d]);
      mn[d] = k < mn[d] ? k : mn[d];
      mx[d] = k > mx[d] ? k : mx[d];
    }
  }
  // wave32 reduction
#pragma unroll
  for (int off = 16; off > 0; off >>= 1) {
#pragma unroll
    for (int d = 0; d < 4; ++d) {
      unsigned a = __shfl_down(mn[d], off, 32); mn[d] = a < mn[d] ? a : mn[d];
      unsigned b = __shfl_down(mx[d], off, 32); mx[d] = b > mx[d] ? b : mx[d];
    }
  }
  if ((threadIdx.x & 31) == 0) {
#pragma unroll
    for (int d = 0; d < 4; ++d) {
      atomicMin(&hdr[HDR_MINKEY + d], mn[d]);
      atomicMax(&hdr[HDR_MAXKEY + d], mx[d]);
    }
  }
}

__global__ void k_setup(int* hdr) {
  if (threadIdx.x != 0) return;
  const float size[4] = {GRIDSZ, GRIDSZ, GRIDSZ, 1.0f};
  float start[4], end[4];
#pragma unroll
  for (int d = 0; d < 4; ++d) {
    start[d] = fdec(((unsigned*)hdr)[HDR_MINKEY + d]);
    end[d]   = fdec(((unsigned*)hdr)[HDR_MAXKEY + d]);
  }
  long long s = 1;
#pragma unroll
  for (int d = 0; d < 4; ++d) {
    hdr[HDR_START + d]  = __float_as_int(start[d]);
    hdr[HDR_STRIDE + d] = (int)s;
    const int dim = (int)floorf((end[d] - start[d]) / size[d]) + 1;
    s *= (long long)dim;
  }
  hdr[HDR_T] = (int)(s < (long long)VOXCAP ? s : (long long)VOXCAP);
}

__global__ void k_hash(const float* __restrict__ pos, const int* __restrict__ batch,
                       const int* __restrict__ hdr, int* __restrict__ hashes,
                       int* __restrict__ occ) {
  const float* hf = (const float*)hdr;
  const float s0 = hf[HDR_START + 0], s1 = hf[HDR_START + 1],
              s2 = hf[HDR_START + 2], s3 = hf[HDR_START + 3];
  const int t1 = hdr[HDR_STRIDE + 1], t2 = hdr[HDR_STRIDE + 2],
            t3 = hdr[HDR_STRIDE + 3];
  const int T = hdr[HDR_T];
  const int stride = gridDim.x * blockDim.x;
  for (int p = blockIdx.x * blockDim.x + threadIdx.x; p < N_PTS; p += stride) {
    const int c0 = (int)floorf((pos[(size_t)p * 3 + 0] - s0) / GRIDSZ);
    const int c1 = (int)floorf((pos[(size_t)p * 3 + 1] - s1) / GRIDSZ);
    const int c2 = (int)floorf((pos[(size_t)p * 3 + 2] - s2) / GRIDSZ);
    const int c3 = (int)floorf((float)batch[p] - s3);
    int h = c0 + c1 * t1 + c2 * t2 + c3 * t3;
    if ((unsigned)h >= (unsigned)T) h = 0;   // defensive clamp
    hashes[p] = h;
    occ[h] = 1;                               // idempotent mark
  }
}

// ---- generic 3-pass exclusive scan over VOXCAP ints ----
__global__ void k_scanA(const int* __restrict__ in, int* __restrict__ bsum) {
  __shared__ int s[SCAN_BLOCK];
  const int t = threadIdx.x, b = blockIdx.x;
  const int* base = in + (size_t)b * SCAN_TILE + t * SCAN_EPT;
  int sum = 0;
#pragma unroll
  for (int i = 0; i < SCAN_EPT; ++i) sum += base[i];
  s[t] = sum; __syncthreads();
  for (int off = SCAN_BLOCK / 2; off > 0; off >>= 1) {
    if (t < off) s[t] += s[t + off];
    __syncthreads();
  }
  if (t == 0) bsum[b] = s[0];
}

__global__ void k_scanB(const int* __restrict__ bsum, int* __restrict__ boff,
                        int* __restrict__ total) {
  __shared__ int s[SCAN_NBLK];
  const int t = threadIdx.x;
  const int v = bsum[t];
  s[t] = v; __syncthreads();
  for (int off = 1; off < SCAN_NBLK; off <<= 1) {
    const int add = (t >= off) ? s[t - off] : 0;
    __syncthreads();
    s[t] += add;
    __syncthreads();
  }
  boff[t] = s[t] - v;                       // exclusive
  if (t == SCAN_NBLK - 1) *total = s[t];
}

__global__ void k_scanC(const int* __restrict__ in, const int* __restrict__ boff,
                        int* __restrict__ outp) {
  __shared__ int s[SCAN_BLOCK];
  const int t = threadIdx.x, b = blockIdx.x;
  const int* base = in + (size_t)b * SCAN_TILE + t * SCAN_EPT;
  int loc[SCAN_EPT]; int sum = 0;
#pragma unroll
  for (int i = 0; i < SCAN_EPT; ++i) { loc[i] = base[i]; sum += loc[i]; }
  s[t] = sum; __syncthreads();
  for (int off = 1; off < SCAN_BLOCK; off <<= 1) {
    const int add = (t >= off) ? s[t - off] : 0;
    __syncthreads();
    s[t] += add;
    __syncthreads();
  }
  int run = boff[b] + s[t] - sum;           // exclusive within block + block offset
  int* pbase = outp + (size_t)b * SCAN_TILE + t * SCAN_EPT;
#pragma unroll
  for (int i = 0; i < SCAN_EPT; ++i) { pbase[i] = run; run += loc[i]; }
}

// relabel hashes -> consecutive cluster ids and count members per cluster
__global__ void k_count(int* __restrict__ hashes, const int* __restrict__ pref,
                        int* __restrict__ counts) {
  const int stride = gridDim.x * blockDim.x;
  for (int p = blockIdx.x * blockDim.x + threadIdx.x; p < N_PTS; p += stride) {
    const int inv = pref[hashes[p]];
    hashes[p] = inv;                         // store cluster id in place
    atomicAdd(&counts[inv], 1);
  }
}

// counting-sort bucket fill: point list grouped by cluster
__global__ void k_order(const int* __restrict__ hashes, const int* __restrict__ cstart,
                        int* __restrict__ cursor, int* __restrict__ order) {
  const int stride = gridDim.x * blockDim.x;
  for (int p = blockIdx.x * blockDim.x + threadIdx.x; p < N_PTS; p += stride) {
    const int inv = hashes[p];
    const int slot = cstart[inv] + atomicAdd(&cursor[inv], 1);
    order[slot] = p;
  }
}

// wave-per-cluster segmented reduce using the Tensor Data Mover in GATHER mode:
// one tensor_load_to_lds fetches up to 8 arbitrary 256B rows of x into this
// wave's LDS slot (TENSORcnt-tracked); the wave then reduces max from LDS and
// writes each output row exactly once.
__global__ void k_cluster(const float* __restrict__ x, const float* __restrict__ pos,
                          const int* __restrict__ batch, const int* __restrict__ order,
                          const int* __restrict__ cstart, const int* __restrict__ counts,
                          float* __restrict__ out) {
  __shared__ float stage[8][8][N_FEAT];   // sole LDS object (offset 0): 8 waves x 8 rows x 256B
  const int wave = threadIdx.x >> 5;
  const int lane = threadIdx.x & 31;
  const int gw = blockIdx.x * (blockDim.x >> 5) + wave;
  const int nw = gridDim.x * (blockDim.x >> 5);
  const unsigned ldsb =
      (unsigned)__builtin_amdgcn_readfirstlane(wave * 8 * N_FEAT * (int)sizeof(float));
  const unsigned long long xa = (unsigned long long)(uintptr_t)x;  // uniform kernarg
  float* posdst = out + (size_t)N_PTS * N_FEAT;
  float* batdst = out + (size_t)N_PTS * 67;
  const float NEGINF = __int_as_float(0xFF800000);

  // D# group0 (constant per wave): count=1 | gather_index_size=32b | gather_mode=1;
  // lds_addr; global_addr (57-bit) in words 2..3; descriptor type=2 in bits 127:126.
  const v4u g0 = { 0xC0000001u, ldsb, (unsigned)xa,
                   (unsigned)(xa >> 32) | 0x80000000u };

  for (int r = gw; r < VOXCAP; r += nw) {
    const int cnt = counts[r];               // uniform across wave
    if (cnt == 0) continue;                  // padding rows stay zero from k_init
    const int base = cstart[r];
    float2 m = make_float2(NEGINF, NEGINF);
    float psum = 0.0f;                       // lanes 0..2 accumulate pos dims
    int pmax = 0;
    for (int c0 = 0; c0 < cnt; c0 += 8) {
      const int nch = (cnt - c0) < 8 ? (cnt - c0) : 8;
      unsigned idx[8];
#pragma unroll
      for (int i = 0; i < 8; ++i) {
        int pi = (i < nch) ? order[base + c0 + i] : 0;
        pi = __builtin_amdgcn_readfirstlane(pi);   // wave-uniform row index
        idx[i] = (unsigned)pi;
        if (i < nch) {
          pmax = pi > pmax ? pi : pmax;
          if (lane < 3) psum += pos[(size_t)pi * 3 + lane];
          __builtin_prefetch(x + (size_t)pi * N_FEAT + lane * 2, 0, 1);
        }
      }
      // D# group1: data_size=4B; tensor_dim0=64; tensor_dim1=2^20 rows;
      // tile_dim0=64; tile_dim1=#valid indices; tensor_dim0_stride=64.
      const v8u g1 = { 0x20000u,
                       64u << 16,
                       0u,
                       (64u << 16) | 0x10u,
                       (unsigned)nch,
                       64u,
                       0u, 0u };
      const v4u g2 = { idx[0], idx[1], idx[2], idx[3] };   // row_index_0..3
      const v4u g3 = { idx[4], idx[5], idx[6], idx[7] };   // row_index_4..7
      asm volatile("tensor_load_to_lds %0, %1, %2, %3"
                   :: "s"(g0), "s"(g1), "s"(g2), "s"(g3)
                   : "memory");
      __builtin_amdgcn_s_wait_tensorcnt(0);
#pragma unroll
      for (int i = 0; i < 8; ++i) {
        if (i < nch) {
          const float2 v =
              *reinterpret_cast<const float2*>(&stage[wave][i][lane * 2]);
          m.x = v.x > m.x ? v.x : m.x;
          m.y = v.y > m.y ? v.y : m.y;
        }
      }
    }
    reinterpret_cast<float2*>(out + (size_t)r * N_FEAT)[lane] = m;
    if (lane < 3)       posdst[(size_t)r * 3 + lane] = psum / (float)cnt;
    else if (lane == 3) batdst[r] = (float)batch[pmax];
  }
}

// ---------------- host launch ----------------
extern "C" void kernel_launch(void* const* d_in, const int* in_sizes, int n_in,
                              void* d_out, int out_size, void* d_ws, size_t ws_size,
                              hipStream_t stream) {
  (void)in_sizes; (void)n_in; (void)out_size; (void)ws_size;
  const float* x     = (const float*)d_in[0];
  const float* pos   = (const float*)d_in[1];
  const int*   batch = (const int*)d_in[2];
  float* out = (float*)d_out;

  int* ws     = (int*)d_ws;            // ~19 MB total
  int* hdr    = ws;                    // 64 ints
  int* hashes = ws + 64;               // N  (voxel hash, then cluster id)
  int* occ    = hashes + N_PTS;        // VOXCAP
  int* pref   = occ + VOXCAP;          // VOXCAP
  int* counts = pref + VOXCAP;         // VOXCAP
  int* cstart = counts + VOXCAP;       // VOXCAP
  int* cursor = cstart + VOXCAP;       // VOXCAP
  int* order  = cursor + VOXCAP;       // N
  int* bsum   = order + N_PTS;         // SCAN_NBLK
  int* boff   = bsum + SCAN_NBLK;      // SCAN_NBLK

  k_init   <<<4096, 256, 0, stream>>>(out, hdr, occ, pref, counts, cstart, cursor);
  k_minmax <<<2048, 256, 0, stream>>>(pos, batch, (unsigned*)hdr);
  k_setup  <<<1, 1, 0, stream>>>(hdr);
  k_hash   <<<4096, 256, 0, stream>>>(pos, batch, hdr, hashes, occ);
  // occupancy -> consecutive cluster ids (hash-sorted relabel)
  k_scanA  <<<SCAN_NBLK, SCAN_BLOCK, 0, stream>>>(occ, bsum);
  k_scanB  <<<1, SCAN_NBLK, 0, stream>>>(bsum, boff, hdr + HDR_K);
  k_scanC  <<<SCAN_NBLK, SCAN_BLOCK, 0, stream>>>(occ, boff, pref);
  // counting sort by cluster id
  k_count  <<<4096, 256, 0, stream>>>(hashes, pref, counts);
  k_scanA  <<<SCAN_NBLK, SCAN_BLOCK, 0, stream>>>(counts, bsum);
  k_scanB  <<<1, SCAN_NBLK, 0, stream>>>(bsum, boff, hdr + HDR_NTOT);
  k_scanC  <<<SCAN_NBLK, SCAN_BLOCK, 0, stream>>>(counts, boff, cstart);
  k_order  <<<4096, 256, 0, stream>>>(hashes, cstart, cursor, order);
  // segmented gather-reduce via TDM gather mode
  k_cluster<<<4096, 256, 0, stream>>>(x, pos, batch, order, cstart, counts, out);
}